// QGMLRegressionTrainer_16569983828320
// MI455X (gfx1250) — compile-verified
//
#include <hip/hip_runtime.h>
#include <hip/hip_bf16.h>

#define NDIM 32     // complex matrix size
#define DDIM 16     // number of A_k matrices / x dim
#define MSZ  64     // real-representation size (2*NDIM)
#define MP   65     // padded LDS row stride (bank-conflict dodge)
#define NSQ  14     // squarings -> M^(2^14) power steps

typedef float v2f __attribute__((ext_vector_type(2)));
typedef float v8f __attribute__((ext_vector_type(8)));

// wave32 max+argmax tree reduction
__device__ __forceinline__ void wave_max_arg(float& v, int& idx) {
#pragma unroll
  for (int m = 16; m >= 1; m >>= 1) {
    float ov = __shfl_xor(v, m, 32);
    int oi = __shfl_xor(idx, m, 32);
    if (ov > v) { v = ov; idx = oi; }
  }
}

__device__ __forceinline__ float wave_max(float v) {
#pragma unroll
  for (int m = 16; m >= 1; m >>= 1) v = fmaxf(v, __shfl_xor(v, m, 32));
  return v;
}

__device__ __forceinline__ float wave_sum(float v) {
#pragma unroll
  for (int m = 16; m >= 1; m >>= 1) v += __shfl_xor(v, m, 32);
  return v;
}

// ---------------------------------------------------------------------------
// Kernel 1: S2 = sum_k A_k @ A_k  (complex 32x32), batch-independent.
// ---------------------------------------------------------------------------
__global__ void s2_kernel(const float* __restrict__ Ar, const float* __restrict__ Ai,
                          float* __restrict__ s2r, float* __restrict__ s2i) {
  int t = blockIdx.x * blockDim.x + threadIdx.x;
  if (t >= NDIM * NDIM) return;
  int i = t / NDIM, l = t % NDIM;
  float sr = 0.f, si = 0.f;
  for (int k = 0; k < DDIM; ++k) {
    const float* ar = Ar + k * NDIM * NDIM;
    const float* ai = Ai + k * NDIM * NDIM;
    for (int j = 0; j < NDIM; ++j) {
      float arij = ar[i * NDIM + j], aiij = ai[i * NDIM + j];
      float arjl = ar[j * NDIM + l], aijl = ai[j * NDIM + l];
      sr += arij * arjl - aiij * aijl;
      si += arij * aijl + aiij * arjl;
    }
  }
  s2r[t] = sr;
  s2i[t] = si;
}

// ---------------------------------------------------------------------------
// Kernel 2: one workgroup (128 thr = 4 waves) per batch element.
//  - assemble real 64x64 rep of H_b in LDS
//  - Gershgorin shift -> PSD M = sigma*I - H
//  - 14x WMMA-f32 squaring (64x64x64 GEMM each) with renormalization
//  - extract dominant column = eigenvector of lambda_min(H)
//  - Rayleigh quotient with B
// ---------------------------------------------------------------------------
__launch_bounds__(128)
__global__ void batch_kernel(const float* __restrict__ Ar, const float* __restrict__ Ai,
                             const float* __restrict__ Br, const float* __restrict__ Bi,
                             const float* __restrict__ x,
                             const float* __restrict__ s2r, const float* __restrict__ s2i,
                             float* __restrict__ out) {
  __shared__ float Ma[MSZ][MP];
  __shared__ float Mb[MSZ][MP];
  __shared__ float xs[DDIM];
  __shared__ float red[MSZ];
  __shared__ float sgm;    // Gershgorin bound
  __shared__ float dmax;   // running normalization scale
  __shared__ int   jstar;  // strongest column index

  const int tid = threadIdx.x;
  const int b = blockIdx.x;
  const int lane = tid & 31;
  const int w    = tid >> 5;    // wave id 0..3 -> output tile-row
  const int lm   = lane & 15;
  const int hi2  = lane >> 4;   // 0: K even pair / M rows +0 ; 1: K odd pair / +8

  if (tid < DDIM) xs[tid] = x[b * DDIM + tid];
  __syncthreads();
  float x2 = 0.f;
  for (int k = 0; k < DDIM; ++k) x2 += xs[k] * xs[k];

  // H = 0.5*S2 - sum_k x_k A_k + 0.5*x2*I  -> real rep [[Hr,-Hi],[Hi,Hr]]
  for (int e = tid; e < NDIM * NDIM; e += blockDim.x) {
    int i = e / NDIM, j = e % NDIM;
    float hr = 0.5f * s2r[e] + (i == j ? 0.5f * x2 : 0.f);
    float hi = 0.5f * s2i[e];
    for (int k = 0; k < DDIM; ++k) {
      hr -= xs[k] * Ar[k * NDIM * NDIM + e];
      hi -= xs[k] * Ai[k * NDIM * NDIM + e];
    }
    Ma[i][j] = hr;
    Ma[i][j + NDIM] = -hi;
    Ma[i + NDIM][j] = hi;
    Ma[i + NDIM][j + NDIM] = hr;
  }
  __syncthreads();

  // Gershgorin upper bound on eigenvalues of the symmetric real rep
  if (tid < MSZ) {
    float g = Ma[tid][tid];
    for (int c = 0; c < MSZ; ++c)
      if (c != tid) g += fabsf(Ma[tid][c]);
    red[tid] = g;
  }
  __syncthreads();
  if (w == 0) {
    float g = fmaxf(red[lane], red[lane + 32]);
    g = wave_max(g);
    if (lane == 0) { sgm = g; dmax = 1.0f; }
  }
  __syncthreads();

  // M = sigma*I - Hrep (PSD; dominant eigenspace == lambda_min(H) space)
  {
    float sg = sgm;
    for (int e = tid; e < MSZ * MSZ; e += blockDim.x) {
      int r = e / MSZ, c = e % MSZ;
      Ma[r][c] = (r == c ? sg : 0.f) - Ma[r][c];
    }
  }
  __syncthreads();

  float(*src)[MP] = Ma;
  float(*dst)[MP] = Mb;
  const int rowa = 16 * w + lm;

  for (int it = 0; it < NSQ; ++it) {
    const float rs = 1.0f / dmax;   // normalize: next = (M/d) @ (M/d)

    v8f acc0 = {}, acc1 = {}, acc2 = {}, acc3 = {};
#pragma unroll
    for (int s = 0; s < 16; ++s) {
      const int k0 = 4 * s + 2 * hi2;
      v2f a;   // A 16x4 frag: lanes 0-15 K={k0,k0+1}, lanes 16-31 K+=2
      a.x = src[rowa][k0] * rs;
      a.y = src[rowa][k0 + 1] * rs;
      v2f b0, b1, b2, b3;  // B 4x16 frags for the 4 column tiles
      b0.x = src[k0][lm] * rs;           b0.y = src[k0 + 1][lm] * rs;
      b1.x = src[k0][16 + lm] * rs;      b1.y = src[k0 + 1][16 + lm] * rs;
      b2.x = src[k0][32 + lm] * rs;      b2.y = src[k0 + 1][32 + lm] * rs;
      b3.x = src[k0][48 + lm] * rs;      b3.y = src[k0 + 1][48 + lm] * rs;
      acc0 = __builtin_amdgcn_wmma_f32_16x16x4_f32(false, a, false, b0, (short)0, acc0, false, false);
      acc1 = __builtin_amdgcn_wmma_f32_16x16x4_f32(false, a, false, b1, (short)0, acc1, false, false);
      acc2 = __builtin_amdgcn_wmma_f32_16x16x4_f32(false, a, false, b2, (short)0, acc2, false, false);
      acc3 = __builtin_amdgcn_wmma_f32_16x16x4_f32(false, a, false, b3, (short)0, acc3, false, false);
    }
#pragma unroll
    for (int v = 0; v < 8; ++v) {
      const int r = 16 * w + v + 8 * hi2;   // C/D layout: lanes16-31 hold M+8
      dst[r][lm]      = acc0[v];
      dst[r][16 + lm] = acc1[v];
      dst[r][32 + lm] = acc2[v];
      dst[r][48 + lm] = acc3[v];
    }
    __syncthreads();

    if (w == 0) {  // PSD: max |entry| is on the diagonal -> scale & best column
      float v1 = dst[lane][lane];
      int i1 = lane;
      float v2 = dst[lane + 32][lane + 32];
      if (v2 > v1) { v1 = v2; i1 = lane + 32; }
      wave_max_arg(v1, i1);
      if (lane == 0) { dmax = fmaxf(v1, 1e-30f); jstar = i1; }
    }
    __syncthreads();
    float(*tmp)[MP] = src; src = dst; dst = tmp;
  }

  // Column jstar of M^K is the target complex eigenvector up to phase:
  // re = col[0:32], im = col[32:64]  (columns >=32 correspond to i*v -- fine).
  if (w == 0) {
    const int j0 = jstar;
    const int i = lane;
    float pr  = src[i][j0];
    float pim = src[i + NDIM][j0];
    float tr = 0.f, ti = 0.f;
    for (int j = 0; j < NDIM; ++j) {
      float br = Br[i * NDIM + j], bi = Bi[i * NDIM + j];
      float qr = src[j][j0], qi = src[j + NDIM][j0];
      tr += br * qr - bi * qi;
      ti += br * qi + bi * qr;
    }
    float nu = wave_sum(pr * tr + pim * ti);   // Re(psi^H B psi)
    float de = wave_sum(pr * pr + pim * pim);  // ||psi||^2
    if (lane == 0) out[b] = nu / de;
  }
}

// ---------------------------------------------------------------------------
extern "C" void kernel_launch(void* const* d_in, const int* in_sizes, int n_in,
                              void* d_out, int out_size, void* d_ws, size_t ws_size,
                              hipStream_t stream) {
  const float* Ar = (const float*)d_in[0];
  const float* Ai = (const float*)d_in[1];
  const float* Br = (const float*)d_in[2];
  const float* Bi = (const float*)d_in[3];
  const float* xp = (const float*)d_in[4];
  float* out = (float*)d_out;

  float* s2r = (float*)d_ws;              // 1024 floats
  float* s2i = s2r + NDIM * NDIM;         // 1024 floats (8 KB total)

  int batch = in_sizes[4] / DDIM;         // 16384

  hipLaunchKernelGGL(s2_kernel, dim3(4), dim3(256), 0, stream, Ar, Ai, s2r, s2i);
  hipLaunchKernelGGL(batch_kernel, dim3(batch), dim3(128), 0, stream,
                     Ar, Ai, Br, Bi, xp, s2r, s2i, out);
}